// HierarchicalBlending_4879082848572
// MI455X (gfx1250) — compile-verified
//
#include <hip/hip_runtime.h>
#include <stdint.h>

typedef __attribute__((ext_vector_type(2))) float v2f;
typedef __attribute__((ext_vector_type(4))) float v4f;
typedef __attribute__((ext_vector_type(8))) float v8f;

#define D 256
#define TILE 16

// ---------------------------------------------------------------------------
// Kernel 1: segment mean over sorted `batch`. One block (256 threads) per
// graph b. Thread 0 binary-searches [start,end) of segment b; then thread t
// accumulates column t over the segment's masked rows (coalesced 1KB/row).
// count = end-start (ALL rows), sum only over low_mask rows — matches ref.
// low_emb is read-once => non-temporal loads to spare L2 for reusable data.
// ---------------------------------------------------------------------------
__global__ void seg_mean_kernel(const float* __restrict__ low_emb,
                                const uint8_t* __restrict__ low_mask,
                                const int* __restrict__ batch,
                                float* __restrict__ g, int N) {
  __shared__ int s_lo, s_hi;
  int b = blockIdx.x;
  if (threadIdx.x == 0) {
    int lo = 0, hi = N;
    while (lo < hi) { int mid = (lo + hi) >> 1; if (batch[mid] < b)     lo = mid + 1; else hi = mid; }
    s_lo = lo;
    hi = N;
    while (lo < hi) { int mid = (lo + hi) >> 1; if (batch[mid] < b + 1) lo = mid + 1; else hi = mid; }
    s_hi = lo;
  }
  __syncthreads();
  int start = s_lo, end = s_hi;
  int t = threadIdx.x;
  float acc = 0.0f;
  for (int i = start; i < end; ++i) {
    if (low_mask[i])                       // skip unmasked rows: halves HBM read
      acc += __builtin_nontemporal_load(&low_emb[(size_t)i * D + t]);
  }
  float cnt = (float)(end - start);
  g[(size_t)b * D + t] = acc / fmaxf(cnt, 1.0f);
}

// ---------------------------------------------------------------------------
// Kernel 2: out[M,D] = A[M,D] @ W^T + bias, optionally zeroing A columns by
// colmask (anchor_mask fused into the A-fragment load). One wave per 16x16
// output tile, V_WMMA_F32_16X16X4_F32 over K=256 (64 WMMA/tile), exact f32.
// EXEC stays all-1s (tile-uniform guard only) as WMMA requires.
// ---------------------------------------------------------------------------
template <bool MASKED>
__global__ void gemm_bias_wmma(const float* __restrict__ A,
                               const float* __restrict__ W,
                               const float* __restrict__ bias,
                               const uint8_t* __restrict__ colmask,
                               float* __restrict__ out, int M) {
  const int lane = threadIdx.x & 31;
  const int wave = threadIdx.x >> 5;
  const int tiles_n = D / TILE;                       // 16
  int tile = blockIdx.x * (blockDim.x >> 5) + wave;
  int tm = tile / tiles_n;
  int tn = tile % tiles_n;
  if (tm * TILE >= M) return;                         // wave-uniform

  const int mn = lane & 15;                           // row for A frag, col for B frag
  const int kk = (lane >> 4) * 2;                     // K sub-offset per half-wave

  const float* Arow = A + (size_t)(tm * TILE + mn) * D;
  const float* Wrow = W + (size_t)(tn * TILE + mn) * D;   // B[k][n] = W[n][k]

  v8f c = {};
  #pragma unroll 8
  for (int k0 = 0; k0 < D; k0 += 4) {
    int k = k0 + kk;
    v2f a, bfr;
    a.x = Arow[k];
    a.y = Arow[k + 1];
    if (MASKED) {
      a.x = colmask[k]     ? a.x : 0.0f;
      a.y = colmask[k + 1] ? a.y : 0.0f;
    }
    bfr.x = Wrow[k];
    bfr.y = Wrow[k + 1];
    // D = A(16x4) x B(4x16) + C ; 8 args: neg_a,A,neg_b,B,c_mod,C,reuse_a,reuse_b
    c = __builtin_amdgcn_wmma_f32_16x16x4_f32(false, a, false, bfr,
                                              (short)0, c, false, false);
  }

  const int col = tn * TILE + mn;
  const float bv = bias[col];
  const int mbase = tm * TILE + (lane >> 4) * 8;      // C/D layout: VGPR v -> row mbase+v
  #pragma unroll
  for (int v = 0; v < 8; ++v) {
    out[(size_t)(mbase + v) * D + col] = c[v] + bv;
  }
}

// ---------------------------------------------------------------------------
// Kernel 3: attn_low[n,:] = per_graph[batch[n],:]. float4-vectorized; the
// 512MB write stream dominates runtime; per_graph (4MB) stays in L2.
// Output is write-once/never-read => non-temporal stores.
// ---------------------------------------------------------------------------
__global__ void gather_kernel(const float* __restrict__ per_graph,
                              const int* __restrict__ batch,
                              float* __restrict__ out, int N) {
  size_t gid = (size_t)blockIdx.x * blockDim.x + threadIdx.x;
  size_t total = (size_t)N * (D / 4);
  if (gid >= total) return;
  int n  = (int)(gid >> 6);                           // D/4 == 64 float4 per row
  int c4 = (int)(gid & 63);
  int b = batch[n];
  const v4f* src = (const v4f*)(per_graph + (size_t)b * D);
  v4f*       dst = (v4f*)(out + (size_t)n * D);
  __builtin_nontemporal_store(src[c4], &dst[c4]);
}

// ---------------------------------------------------------------------------
extern "C" void kernel_launch(void* const* d_in, const int* in_sizes, int n_in,
                              void* d_out, int out_size, void* d_ws, size_t ws_size,
                              hipStream_t stream) {
  const float*   high_emb    = (const float*)d_in[0];
  const float*   low_emb     = (const float*)d_in[1];
  const int*     batch       = (const int*)d_in[2];
  const uint8_t* anchor_mask = (const uint8_t*)d_in[3];
  const uint8_t* low_mask    = (const uint8_t*)d_in[4];
  const float*   Wv_high     = (const float*)d_in[5];
  const float*   bv_high     = (const float*)d_in[6];
  const float*   Wo_high     = (const float*)d_in[7];
  const float*   bo_high     = (const float*)d_in[8];
  const float*   Wv_low      = (const float*)d_in[9];
  const float*   bv_low      = (const float*)d_in[10];
  const float*   Wo_low      = (const float*)d_in[11];
  const float*   bo_low      = (const float*)d_in[12];

  const int B = in_sizes[0] / D;        // 4096
  const int N = in_sizes[2];            // 500000

  // workspace: 4 x B*D f32 (16 MB total)
  float* g  = (float*)d_ws;
  float* h1 = g  + (size_t)B * D;
  float* p1 = h1 + (size_t)B * D;
  float* pg = p1 + (size_t)B * D;

  float* attn_high = (float*)d_out;             // [B,D]
  float* attn_low  = attn_high + (size_t)B * D; // [N,D]

  // 1) segment mean -> g
  seg_mean_kernel<<<B, 256, 0, stream>>>(low_emb, low_mask, batch, g, N);

  // 2) GEMMs (one wave per 16x16 tile, 8 waves/block)
  const int tiles  = (B / TILE) * (D / TILE);   // 4096
  const int blocks = tiles / 8;                 // 512
  gemm_bias_wmma<false><<<blocks, 256, 0, stream>>>(g,        Wv_high, bv_high, nullptr,     h1,        B);
  gemm_bias_wmma<false><<<blocks, 256, 0, stream>>>(h1,       Wo_high, bo_high, nullptr,     attn_high, B);
  gemm_bias_wmma<true ><<<blocks, 256, 0, stream>>>(high_emb, Wv_low,  bv_low,  anchor_mask, p1,        B);
  gemm_bias_wmma<false><<<blocks, 256, 0, stream>>>(p1,       Wo_low,  bo_low,  nullptr,     pg,        B);

  // 3) gather -> attn_low
  size_t total4  = (size_t)N * (D / 4);
  int    gblocks = (int)((total4 + 255) / 256);
  gather_kernel<<<gblocks, 256, 0, stream>>>(pg, batch, attn_low, N);
}